// AttnDecoderRNN_14061722928005
// MI455X (gfx1250) — compile-verified
//
#include <hip/hip_runtime.h>
#include <hip/hip_bf16.h>
#include <math.h>

// ---------------------------------------------------------------------------
// AttnDecoderRNN single step, MI455X (gfx1250, wave32).
// H=1024, V=50257, L=50, ENC=2H=2048.
// Memory-bound (~257MB fp32 weights @ 23.3TB/s ~= 11us floor). All GEMVs are
// coalesced column-per-thread streamers with split-K for occupancy (partials
// reduced in a fixed order -> deterministic). The [50,2048]x[2048,1024]
// attention GEMM uses V_WMMA_F32_16X16X4_F32. Default RT temporal hints so
// the 192MB L2 retains most weights across graph replays.
// ---------------------------------------------------------------------------

#define Hs   1024
#define Vs   50257
#define Ls   50
#define ENCs 2048

typedef float v2f __attribute__((ext_vector_type(2)));
typedef float v8f __attribute__((ext_vector_type(8)));

// ---------------------------------------------------------------------------
// Split-K streaming GEMV, stage 1:
//   partial[chunk][j] = sum_{k in chunk} vec[k] * W[k*N + j]
// grid = (ceil(N/256), kchunks). Chunk of vec staged in LDS. W rows are read
// fully coalesced (thread j -> consecutive addresses within a row).
// ---------------------------------------------------------------------------
__global__ void gemv_splitk_kernel(const float* __restrict__ vec, int K, int kchunks,
                                   const float* __restrict__ W, int N, int Npad,
                                   float* __restrict__ partial)
{
    __shared__ float v[256];
    const int chunkLen = K / kchunks;         // always <= 256 here
    const int chunk    = blockIdx.y;
    const int k0       = chunk * chunkLen;
    for (int i = threadIdx.x; i < chunkLen; i += blockDim.x) v[i] = vec[k0 + i];
    __syncthreads();

    const int j = blockIdx.x * blockDim.x + threadIdx.x;
    if (j >= N) return;

    const float* __restrict__ Wp = W + (size_t)k0 * (size_t)N + (size_t)j;
    float acc = 0.0f;
#pragma unroll 8
    for (int k = 0; k < chunkLen; ++k)
        acc = fmaf(v[k], Wp[(size_t)k * (size_t)N], acc);

    partial[(size_t)chunk * (size_t)Npad + (size_t)j] = acc;
}

// Split-K GEMV stage 2: fixed-order reduction (deterministic) + bias + relu.
__global__ void gemv_reduce_kernel(const float* __restrict__ partial,
                                   int kchunks, int N, int Npad,
                                   const float* __restrict__ bias,
                                   float* __restrict__ out, int do_relu)
{
    const int j = blockIdx.x * blockDim.x + threadIdx.x;
    if (j >= N) return;
    float acc = (bias != nullptr) ? bias[j] : 0.0f;
    for (int c = 0; c < kchunks; ++c)
        acc += partial[(size_t)c * (size_t)Npad + (size_t)j];
    if (do_relu) acc = fmaxf(acc, 0.0f);
    out[j] = acc;
}

// ---------------------------------------------------------------------------
// encpart = enc[50,2048] @ W2[2048,1024]  via V_WMMA_F32_16X16X4_F32.
// One wave (32 threads) per N-tile of 16 columns; wave holds 4 M-tile
// accumulators (rows 0..63, rows >=50 masked to zero by value-select so EXEC
// stays all-ones across WMMA). B weight fragment is loaded once per k-step
// and reused by all 4 WMMAs.
//
// Fragment layouts (ISA 7.12.2, wave32):
//   A 16x4 f32 : a.x = A[lane&15][k0 + (lane<16?0:2)], a.y = +1
//   B 4x16 f32 : b.x = B[k0 + (lane<16?0:2)][ncol],    b.y = +1
//   C 16x16 f32: acc[i] = C[i + (lane<16?0:8)][lane&15]
// ---------------------------------------------------------------------------
__global__ void enc_wmma_kernel(const float* __restrict__ enc,   // [50,2048]
                                const float* __restrict__ W2,    // [2048,1024]
                                float* __restrict__ encp)        // [64,1024]
{
    const int lane  = threadIdx.x & 31;
    const int ntile = blockIdx.x;               // 0..63
    const int col   = ntile * 16 + (lane & 15);
    const int ksel  = (lane >> 4) << 1;         // 0 for lanes 0-15, 2 for 16-31

    v8f acc0 = {}, acc1 = {}, acc2 = {}, acc3 = {};

    for (int k0 = 0; k0 < ENCs; k0 += 4) {
        const int kb = k0 + ksel;
        v2f b;
        b.x = W2[(size_t)kb       * Hs + col];
        b.y = W2[(size_t)(kb + 1) * Hs + col];

        v2f a;
        int  row;
        int  rc;
        float msk;

        row = 0 * 16 + (lane & 15);
        rc  = row < Ls ? row : (Ls - 1);
        msk = row < Ls ? 1.0f : 0.0f;
        a.x = enc[(size_t)rc * ENCs + kb] * msk;
        a.y = enc[(size_t)rc * ENCs + kb + 1] * msk;
        acc0 = __builtin_amdgcn_wmma_f32_16x16x4_f32(false, a, false, b,
                                                     (short)0, acc0, false, false);

        row = 1 * 16 + (lane & 15);
        rc  = row < Ls ? row : (Ls - 1);
        msk = row < Ls ? 1.0f : 0.0f;
        a.x = enc[(size_t)rc * ENCs + kb] * msk;
        a.y = enc[(size_t)rc * ENCs + kb + 1] * msk;
        acc1 = __builtin_amdgcn_wmma_f32_16x16x4_f32(false, a, false, b,
                                                     (short)0, acc1, false, false);

        row = 2 * 16 + (lane & 15);
        rc  = row < Ls ? row : (Ls - 1);
        msk = row < Ls ? 1.0f : 0.0f;
        a.x = enc[(size_t)rc * ENCs + kb] * msk;
        a.y = enc[(size_t)rc * ENCs + kb + 1] * msk;
        acc2 = __builtin_amdgcn_wmma_f32_16x16x4_f32(false, a, false, b,
                                                     (short)0, acc2, false, false);

        row = 3 * 16 + (lane & 15);
        rc  = row < Ls ? row : (Ls - 1);
        msk = row < Ls ? 1.0f : 0.0f;
        a.x = enc[(size_t)rc * ENCs + kb] * msk;
        a.y = enc[(size_t)rc * ENCs + kb + 1] * msk;
        acc3 = __builtin_amdgcn_wmma_f32_16x16x4_f32(false, a, false, b,
                                                     (short)0, acc3, false, false);
    }

    // Writeback (divergent predication is fine here: no more WMMAs follow).
    const int mbase = (lane >> 4) << 3;  // 0 or 8
    const int ccol  = ntile * 16 + (lane & 15);
#pragma unroll
    for (int i = 0; i < 8; ++i) {
        int r0 = 0 * 16 + i + mbase;
        int r1 = 1 * 16 + i + mbase;
        int r2 = 2 * 16 + i + mbase;
        int r3 = 3 * 16 + i + mbase;
        if (r0 < Ls) encp[(size_t)r0 * Hs + ccol] = acc0[i];
        if (r1 < Ls) encp[(size_t)r1 * Hs + ccol] = acc1[i];
        if (r2 < Ls) encp[(size_t)r2 * Hs + ccol] = acc2[i];
        if (r3 < Ls) encp[(size_t)r3 * Hs + ccol] = acc3[i];
    }
}

// e[l] = sum_n relu(hpart[n] + encp[l][n] + wb[n]) * vW[n]  + vb
__global__ void escore_kernel(const float* __restrict__ hpart,
                              const float* __restrict__ encp,
                              const float* __restrict__ wb,
                              const float* __restrict__ vW,
                              const float* __restrict__ vb,
                              float* __restrict__ e)
{
    __shared__ float red[256];
    const int l = blockIdx.x;
    float acc = 0.0f;
    for (int n = threadIdx.x; n < Hs; n += 256) {
        float t = fmaxf(hpart[n] + encp[(size_t)l * Hs + n] + wb[n], 0.0f);
        acc = fmaf(t, vW[n], acc);
    }
    red[threadIdx.x] = acc;
    __syncthreads();
    for (int s = 128; s > 0; s >>= 1) {
        if (threadIdx.x < s) red[threadIdx.x] += red[threadIdx.x + s];
        __syncthreads();
    }
    if (threadIdx.x == 0) e[l] = red[0] + vb[0];
}

// softmax over 50 scores; writes attn weights to scratch and to output slot.
__global__ void softmax50_kernel(const float* __restrict__ e,
                                 float* __restrict__ aw,
                                 float* __restrict__ out_aw)
{
    __shared__ float red[64];
    const int t = threadIdx.x;
    float v = (t < Ls) ? e[t] : -INFINITY;
    red[t] = v;
    __syncthreads();
    for (int s = 32; s > 0; s >>= 1) {
        if (t < s) red[t] = fmaxf(red[t], red[t + s]);
        __syncthreads();
    }
    float mx = red[0];
    __syncthreads();
    float ex = (t < Ls) ? __expf(v - mx) : 0.0f;
    red[t] = ex;
    __syncthreads();
    for (int s = 32; s > 0; s >>= 1) {
        if (t < s) red[t] += red[t + s];
        __syncthreads();
    }
    float w = ex / red[0];
    if (t < Ls) { aw[t] = w; out_aw[t] = w; }
}

// attn_applied[c] = sum_l aw[l] * enc[l][c]   (c in [0,2048))
__global__ void attnapply_kernel(const float* __restrict__ aw,
                                 const float* __restrict__ enc,
                                 float* __restrict__ out)
{
    __shared__ float w[Ls];
    if (threadIdx.x < Ls) w[threadIdx.x] = aw[threadIdx.x];
    __syncthreads();
    const int c = blockIdx.x * 256 + threadIdx.x;
    float acc = 0.0f;
#pragma unroll 5
    for (int l = 0; l < Ls; ++l)
        acc = fmaf(w[l], enc[(size_t)l * ENCs + c], acc);
    out[c] = acc;
}

// xcat = concat(emb[idx], attn_applied)  -> [3072]
__global__ void xcat_kernel(const int* __restrict__ idx,
                            const float* __restrict__ emb,
                            const float* __restrict__ attn,
                            float* __restrict__ xcat)
{
    const int k = blockIdx.x * 256 + threadIdx.x;
    if (k >= 3 * Hs) return;
    const long long row = (long long)idx[0];
    xcat[k] = (k < Hs) ? emb[row * (long long)Hs + k] : attn[k - Hs];
}

// GRU gate fusion (PyTorch order r,z,n); writes h_new to scratch + output.
__global__ void gru_gate_kernel(const float* __restrict__ gi,
                                const float* __restrict__ gh,
                                const float* __restrict__ h,
                                float* __restrict__ hnew,
                                float* __restrict__ out_h)
{
    const int j = blockIdx.x * 256 + threadIdx.x;
    if (j >= Hs) return;
    float r  = 1.0f / (1.0f + __expf(-(gi[j] + gh[j])));
    float z  = 1.0f / (1.0f + __expf(-(gi[Hs + j] + gh[Hs + j])));
    float n  = tanhf(gi[2 * Hs + j] + r * gh[2 * Hs + j]);
    float hn = (1.0f - z) * n + z * h[j];
    hnew[j]  = hn;
    out_h[j] = hn;
}

// Single-block log-sum-exp over V logits.
__global__ void lse_kernel(const float* __restrict__ logits,
                           float* __restrict__ lse)
{
    __shared__ float red[1024];
    float m = -INFINITY;
    for (int j = threadIdx.x; j < Vs; j += 1024) m = fmaxf(m, logits[j]);
    red[threadIdx.x] = m;
    __syncthreads();
    for (int s = 512; s > 0; s >>= 1) {
        if (threadIdx.x < s) red[threadIdx.x] = fmaxf(red[threadIdx.x], red[threadIdx.x + s]);
        __syncthreads();
    }
    const float mx = red[0];
    __syncthreads();
    float sum = 0.0f;
    for (int j = threadIdx.x; j < Vs; j += 1024) sum += __expf(logits[j] - mx);
    red[threadIdx.x] = sum;
    __syncthreads();
    for (int s = 512; s > 0; s >>= 1) {
        if (threadIdx.x < s) red[threadIdx.x] += red[threadIdx.x + s];
        __syncthreads();
    }
    if (threadIdx.x == 0) lse[0] = mx + __logf(red[0]);
}

__global__ void logprob_kernel(const float* __restrict__ logits,
                               const float* __restrict__ lse,
                               float* __restrict__ out)
{
    const int j = blockIdx.x * 256 + threadIdx.x;
    if (j < Vs) out[j] = logits[j] - lse[0];
}

// ---------------------------------------------------------------------------
// Launch
// ---------------------------------------------------------------------------
static inline void gemv_splitk(const float* vec, int K, int kchunks,
                               const float* W, int N, int Npad,
                               const float* bias, float* out, int do_relu,
                               float* partial, hipStream_t stream)
{
    dim3 grid1((N + 255) / 256, kchunks);
    gemv_splitk_kernel<<<grid1, 256, 0, stream>>>(vec, K, kchunks, W, N, Npad, partial);
    gemv_reduce_kernel<<<(N + 255) / 256, 256, 0, stream>>>(
        partial, kchunks, N, Npad, bias, out, do_relu);
}

extern "C" void kernel_launch(void* const* d_in, const int* in_sizes, int n_in,
                              void* d_out, int out_size, void* d_ws, size_t ws_size,
                              hipStream_t stream)
{
    (void)in_sizes; (void)n_in; (void)out_size; (void)ws_size;

    const int*   input_idx = (const int*)  d_in[0];
    const float* hidden    = (const float*)d_in[1];   // [H]
    const float* enc       = (const float*)d_in[2];   // [L, 2H]
    const float* emb       = (const float*)d_in[3];   // [V, H]
    const float* attn_w_W  = (const float*)d_in[4];   // [3H, H]
    const float* attn_w_b  = (const float*)d_in[5];   // [H]
    const float* attn_v_W  = (const float*)d_in[6];   // [H]
    const float* attn_v_b  = (const float*)d_in[7];   // [1]
    const float* comb_W    = (const float*)d_in[8];   // [3H, H]
    const float* comb_b    = (const float*)d_in[9];   // [H]
    const float* gru_wih   = (const float*)d_in[10];  // [H, 3H]
    const float* gru_bih   = (const float*)d_in[11];  // [3H]
    const float* gru_whh   = (const float*)d_in[12];  // [H, 3H]
    const float* gru_bhh   = (const float*)d_in[13];  // [3H]
    const float* out_W     = (const float*)d_in[14];  // [H, V]
    const float* out_b     = (const float*)d_in[15];  // [V]

    float* out = (float*)d_out;     // [V logprobs][H h_new][L attn_weights]
    float* out_logp = out;
    float* out_h    = out + Vs;
    float* out_aw   = out + Vs + Hs;

    const int VsPad = (Vs + 31) & ~31;    // 50272

    // Workspace layout (floats)
    float* ws        = (float*)d_ws;
    float* ws_hpart  = ws;                   // 1024
    float* ws_encp   = ws_hpart + Hs;        // 64*1024
    float* ws_e      = ws_encp + 64 * Hs;    // 64
    float* ws_aw     = ws_e + 64;            // 64
    float* ws_attn   = ws_aw + 64;           // 2048
    float* ws_xcat   = ws_attn + ENCs;       // 3072
    float* ws_x      = ws_xcat + 3 * Hs;     // 1024
    float* ws_gi     = ws_x + Hs;            // 3072
    float* ws_gh     = ws_gi + 3 * Hs;       // 3072
    float* ws_hnew   = ws_gh + 3 * Hs;       // 1024
    float* ws_logit  = ws_hnew + Hs;         // VsPad
    float* ws_lse    = ws_logit + VsPad;     // 32
    float* ws_part   = ws_lse + 32;          // 8*VsPad (reused by every GEMV)

    // 1) hpart = h @ attn_w_W[0:H,:]   (broadcast row of att_cat hoisted out)
    gemv_splitk(hidden, Hs, 16, attn_w_W, Hs, Hs,
                nullptr, ws_hpart, 0, ws_part, stream);

    // 2) encpart = enc @ attn_w_W[H:3H,:]   -- WMMA f32 GEMM
    enc_wmma_kernel<<<Hs / 16, 32, 0, stream>>>(
        enc, attn_w_W + (size_t)Hs * Hs, ws_encp);

    // 3) e[l] = relu(hpart + encpart[l] + b) . attn_v_W + attn_v_b
    escore_kernel<<<Ls, 256, 0, stream>>>(
        ws_hpart, ws_encp, attn_w_b, attn_v_W, attn_v_b, ws_e);

    // 4) attn_weights = softmax(e)
    softmax50_kernel<<<1, 64, 0, stream>>>(ws_e, ws_aw, out_aw);

    // 5) attn_applied = aw @ enc
    attnapply_kernel<<<ENCs / 256, 256, 0, stream>>>(ws_aw, enc, ws_attn);

    // 6) xcat = [emb[idx], attn_applied];  x = relu(xcat @ comb_W + comb_b)
    xcat_kernel<<<(3 * Hs + 255) / 256, 256, 0, stream>>>(
        input_idx, emb, ws_attn, ws_xcat);
    gemv_splitk(ws_xcat, 3 * Hs, 16, comb_W, Hs, Hs,
                comb_b, ws_x, 1, ws_part, stream);

    // 7) GRU pre-activations (K split x16 -> ~1.5K waves each)
    gemv_splitk(ws_x, Hs, 16, gru_wih, 3 * Hs, 3 * Hs,
                gru_bih, ws_gi, 0, ws_part, stream);
    gemv_splitk(hidden, Hs, 16, gru_whh, 3 * Hs, 3 * Hs,
                gru_bhh, ws_gh, 0, ws_part, stream);

    // 8) gates -> h_new
    gru_gate_kernel<<<(Hs + 255) / 256, 256, 0, stream>>>(
        ws_gi, ws_gh, hidden, ws_hnew, out_h);

    // 9) logits = h_new @ out_W + out_b  (206MB sweep; K split x8 -> ~12.6K waves)
    gemv_splitk(ws_hnew, Hs, 8, out_W, Vs, VsPad,
                out_b, ws_logit, 0, ws_part, stream);

    // 10) log_softmax
    lse_kernel<<<1, 1024, 0, stream>>>(ws_logit, ws_lse);
    logprob_kernel<<<(Vs + 255) / 256, 256, 0, stream>>>(ws_logit, ws_lse, out_logp);
}